// Separator_57647051047620
// MI455X (gfx1250) — compile-verified
//
#include <hip/hip_runtime.h>
#include <math.h>

// ---------------------------------------------------------------------------
// Open-Unmix style separator for MI455X (gfx1250, wave32).
// STFT / Wiener(1 EM pass) / iSTFT.  FFT-4096 = 3 x radix-16 Stockham stages,
// each radix-16 butterfly batch done as 16x16x16 f32 matmul via chained
// V_WMMA_F32_16X16X4_F32 (full fp32 precision; problem is HBM-bound at
// ~0.8 GB total traffic / 23.3 TB/s, so fp32 matrix path costs nothing).
// Global->LDS traffic uses CDNA5 async-to-LDS DMA (ASYNCcnt) instead of the
// CDNA4-style VGPR round trip.
// ---------------------------------------------------------------------------

typedef float v2f __attribute__((ext_vector_type(2)));
typedef float v8f __attribute__((ext_vector_type(8)));

#define TWO_PI_F 6.28318530717958647692f
#define NFFT     4096
#define HOP      1024
#define NBINS    2049
#define NFRAMES  600
#define TLEN     613376
#define NWIN     300          // wiener chunk length (frames)
#define NSRC     4
#define NCH      2
#define NBATCH   2
#define EPS_F    1e-10f
#define SQEPS_F  1e-5f        // sqrt(1e-10)

// ---------------- complex helpers --------------------------------------------
__device__ __forceinline__ float2 cmul(float2 a, float2 b) {
  return make_float2(a.x*b.x - a.y*b.y, a.x*b.y + a.y*b.x);
}
__device__ __forceinline__ float2 cadd(float2 a, float2 b) {
  return make_float2(a.x + b.x, a.y + b.y);
}
__device__ __forceinline__ float2 csub(float2 a, float2 b) {
  return make_float2(a.x - b.x, a.y - b.y);
}
__device__ __forceinline__ float2 cscale(float2 a, float s) {
  return make_float2(a.x*s, a.y*s);
}
__device__ __forceinline__ float hann_w(int i) {
  return 0.5f - 0.5f*__cosf((TWO_PI_F/(float)NFFT)*(float)i);
}

// ---------------- CDNA5 async global->LDS DMA --------------------------------
// Per-lane gather directly into LDS; tracked by ASYNCcnt.  The LDS operand is
// the 32-bit LDS byte offset (low 32 bits of the generic shared address).
__device__ __forceinline__ void async_gload_b32(unsigned lds_off, const void* gptr) {
  asm volatile("global_load_async_to_lds_b32 %0, %1, off"
               :: "v"(lds_off), "v"(gptr) : "memory");
}
__device__ __forceinline__ void async_gload_b64(unsigned lds_off, const void* gptr) {
  asm volatile("global_load_async_to_lds_b64 %0, %1, off"
               :: "v"(lds_off), "v"(gptr) : "memory");
}
__device__ __forceinline__ void wait_async0() {
  asm volatile("s_wait_asynccnt 0" ::: "memory");
}
__device__ __forceinline__ unsigned lds_offset_of(const void* shared_ptr) {
  return (unsigned)(size_t)shared_ptr;   // generic LDS addr carries offset in [31:0]
}

// ---------------- WMMA f32 16x16x4: D = A(16x4) * B(4x16) + C ----------------
__device__ __forceinline__ v8f wmma4(v2f a, v2f b, v8f c) {
  // (neg_a, A, neg_b, B, c_mod, C, reuse_a, reuse_b)
  return __builtin_amdgcn_wmma_f32_16x16x4_f32(false, a, false, b, (short)0, c,
                                               false, false);
}

// Constant DFT-16 matrix fragments in WMMA A-layout:
// lane L: half h = L>>4, row m = L&15; chunk c covers K = 4c..4c+3 with
// VGPR0 <- K = 4c+2h, VGPR1 <- K = 4c+2h+1  (ISA 16x4 f32 A layout).
struct DftFrags { v2f Fr[4]; v2f Fi[4]; v2f Fin[4]; };

__device__ __forceinline__ DftFrags make_dft_frags() {
  DftFrags fg;
  const int lane = threadIdx.x & 31;
  const int h = lane >> 4;
  const int m = lane & 15;
#pragma unroll
  for (int c = 0; c < 4; ++c) {
    const int r0 = 4*c + 2*h;
    float s0, c0, s1, c1;
    __sincosf((TWO_PI_F/16.f)*(float)(m*r0),     &s0, &c0);
    __sincosf((TWO_PI_F/16.f)*(float)(m*(r0+1)), &s1, &c1);
    fg.Fr[c]  = (v2f){ c0,  c1};   // Re W16^{mr} = cos
    fg.Fi[c]  = (v2f){-s0, -s1};   // Im W16^{mr} = -sin
    fg.Fin[c] = (v2f){ s0,  s1};   // pre-negated Fi (f32 WMMA has no A-neg)
  }
  return fg;
}

// 4096-pt complex FFT, 3 Stockham radix-16 stages, 128 threads (4 waves).
// Input in bufA, result lands in bufB (natural order).  Caller must
// __syncthreads() after filling bufA.
__device__ void fft4096(float2* bufA, float2* bufB, const DftFrags& fg) {
  const int lane = threadIdx.x & 31;
  const int wave = threadIdx.x >> 5;
  const int h    = lane >> 4;
  const int n16  = lane & 15;

#pragma unroll
  for (int st = 0; st < 3; ++st) {
    const float2* in  = (st == 1) ? bufB : bufA;   // A->B, B->A, A->B
    float2*       out = (st == 1) ? bufA : bufB;

    for (int bt = wave; bt < 16; bt += 4) {        // 4 waves x 4 batches
      const int col = 16*bt + n16;                 // read base, stride 256
      v8f Dr = {0.f,0.f,0.f,0.f,0.f,0.f,0.f,0.f};
      v8f Di = {0.f,0.f,0.f,0.f,0.f,0.f,0.f,0.f};
#pragma unroll
      for (int c = 0; c < 4; ++c) {
        const int r0 = 4*c + 2*h;
        const float2 v0 = in[col + 256*r0];
        const float2 v1 = in[col + 256*(r0+1)];
        const v2f Br = {v0.x, v1.x};
        const v2f Bi = {v0.y, v1.y};
        Dr = wmma4(fg.Fr[c],  Br, Dr);   // Re += Fr*Br
        Dr = wmma4(fg.Fin[c], Bi, Dr);   // Re += (-Fi)*Bi
        Di = wmma4(fg.Fr[c],  Bi, Di);   // Im += Fr*Bi
        Di = wmma4(fg.Fi[c],  Br, Di);   // Im += Fi*Br
      }
      // twiddle + scatter (Stockham natural-order write)
      float pang; int wbase, wstride;
      if (st == 0) {
        pang = -(TWO_PI_F/4096.f)*(float)col; wbase = 16*col; wstride = 1;
      } else if (st == 1) {
        const int P = col >> 4;
        pang = -(TWO_PI_F/256.f)*(float)P; wbase = (col & 15) + 256*P; wstride = 16;
      } else {
        pang = 0.f; wbase = col; wstride = 256;
      }
#pragma unroll
      for (int j = 0; j < 8; ++j) {
        const int k = j + 8*h;                     // D layout: M = j + 8h
        float s, cn;
        __sincosf(pang*(float)k, &s, &cn);
        out[wbase + wstride*k] =
            make_float2(Dr[j]*cn - Di[j]*s, Dr[j]*s + Di[j]*cn);
      }
    }
    __syncthreads();
  }
}

// ---------------- forward STFT ----------------------------------------------
__global__ __launch_bounds__(128)
void kfft_fwd(const float* __restrict__ audio, float2* __restrict__ mix) {
  __shared__ float2 bufA[NFFT];
  __shared__ float2 bufB[NFFT];
  const int fid = blockIdx.x;            // (n*NCH + c)*NFRAMES + f
  const int f   = fid % NFRAMES;
  const int nc  = fid / NFRAMES;
  const float* src = audio + (size_t)nc * TLEN;

  const DftFrags fg = make_dft_frags();

  // Async per-lane gather of the reflect-padded frame straight into LDS
  // (staged in bufB; it is consumed before stage-0 writes it).
  float* stage = (float*)bufB;
  const unsigned sbase = lds_offset_of(stage);
  for (int i = threadIdx.x; i < NFFT; i += 128) {
    int t = f*HOP + i - (NFFT/2);        // reflect pad (mode='reflect')
    if (t < 0)      t = -t;
    if (t >= TLEN)  t = 2*TLEN - 2 - t;
    async_gload_b32(sbase + 4u*(unsigned)i, src + t);
    int tn = t + HOP; if (tn >= TLEN) tn = TLEN - 1;
    __builtin_prefetch(&src[tn], 0, 1);  // global_prefetch for next frame
  }
  wait_async0();
  __syncthreads();

  for (int i = threadIdx.x; i < NFFT; i += 128)
    bufA[i] = make_float2(stage[i]*hann_w(i), 0.f);
  __syncthreads();

  fft4096(bufA, bufB, fg);

  float2* dst = mix + (size_t)fid * NBINS;
  for (int i = threadIdx.x; i < NBINS; i += 128) dst[i] = bufB[i];
}

// ---------------- inverse STFT frame (irfft * window) ------------------------
__global__ __launch_bounds__(128)
void kfft_inv(const float2* __restrict__ ybuf, float* __restrict__ frames) {
  __shared__ float2 bufA[NFFT];
  __shared__ float2 bufB[NFFT];
  const int fid = blockIdx.x;            // ((n*NSRC+s)*NCH+c)*NFRAMES + f
  const float2* spec = ybuf + (size_t)fid * NBINS;

  const DftFrags fg = make_dft_frags();

  // Async DMA the 2049 complex bins into LDS staging (bufB).
  float2* stage = bufB;
  const unsigned sbase = lds_offset_of(stage);
  for (int i = threadIdx.x; i < NBINS; i += 128)
    async_gload_b64(sbase + 8u*(unsigned)i, spec + i);
  wait_async0();
  __syncthreads();

  // bufA = conj(hermitian_extend(spec)):  ifft(x) = conj(fft(conj(x)))/N
  for (int i = threadIdx.x; i < NFFT; i += 128) {
    if (i <= NFFT/2) { const float2 v = stage[i];        bufA[i] = make_float2(v.x, -v.y); }
    else             { const float2 v = stage[NFFT - i]; bufA[i] = v; }
  }
  __syncthreads();

  fft4096(bufA, bufB, fg);

  float* dst = frames + (size_t)fid * NFFT;
  const float inv_n = 1.f / (float)NFFT;
  for (int i = threadIdx.x; i < NFFT; i += 128)
    dst[i] = bufB[i].x * inv_n * hann_w(i);   // real part of conj(out)/N * win
}

// ---------------- chunk max |x|^2 --------------------------------------------
__global__ void kinit(float* __restrict__ maxbuf) {
  if (threadIdx.x < NBATCH*2) maxbuf[threadIdx.x] = 0.f;
}

__global__ void kmax(const float2* __restrict__ mix, float* __restrict__ maxbuf) {
  __shared__ float red[256];
  const int nchunk = blockIdx.y;               // n*2 + chunk
  const int n = nchunk >> 1, ch = nchunk & 1;
  const int e = blockIdx.x*256 + threadIdx.x;
  const int per = NWIN * NBINS;
  float v = 0.f;
  if (e < NCH*per) {
    const int c = e / per;
    const int r = e % per;
    const int t = r / NBINS;
    const int b = r % NBINS;
    const float2 x =
        mix[(((size_t)(n*NCH + c))*NFRAMES + (ch*NWIN + t))*NBINS + b];
    v = x.x*x.x + x.y*x.y;
  }
  red[threadIdx.x] = v;
  __syncthreads();
  for (int s = 128; s > 0; s >>= 1) {
    if (threadIdx.x < s)
      red[threadIdx.x] = fmaxf(red[threadIdx.x], red[threadIdx.x + s]);
    __syncthreads();
  }
  if (threadIdx.x == 0)
    atomicMax((unsigned int*)&maxbuf[nchunk], __float_as_uint(red[0]));
}

// ---------------- per-bin EM statistics --------------------------------------
// y0 = mask*x  =>  R[b,c,d,s] = m2*Gram_cd / (eps + m2 * sum_t psd)  with
// Gram_cd = sum_t x_c conj(x_d) / ma^2  (source-independent).
__global__ void kstats(const float2* __restrict__ mix, const float* __restrict__ W,
                       const float* __restrict__ maxbuf, float2* __restrict__ Rbuf) {
  const int nchunk = blockIdx.y;
  const int n = nchunk >> 1, ch = nchunk & 1;
  const int b = blockIdx.x*256 + threadIdx.x;
  if (b >= NBINS) return;

  const float ma   = fmaxf(1.f, sqrtf(maxbuf[nchunk]) * 0.1f);
  const float inv2 = 1.f/(ma*ma);
  const float2* m0 = mix + (((size_t)(n*NCH + 0))*NFRAMES + ch*NWIN)*NBINS;
  const float2* m1 = mix + (((size_t)(n*NCH + 1))*NFRAMES + ch*NWIN)*NBINS;

  float g00 = 0.f, g11 = 0.f, g01r = 0.f, g01i = 0.f;
  for (int t = 0; t < NWIN; ++t) {
    const float2 x0 = m0[(size_t)t*NBINS + b];
    const float2 x1 = m1[(size_t)t*NBINS + b];
    g00  += x0.x*x0.x + x0.y*x0.y;
    g11  += x1.x*x1.x + x1.y*x1.y;
    g01r += x0.x*x1.x + x0.y*x1.y;       // Re(x0 * conj(x1))
    g01i += x0.y*x1.x - x0.x*x1.y;       // Im(x0 * conj(x1))
  }
  g00 *= inv2; g11 *= inv2; g01r *= inv2; g01i *= inv2;
  const float P = 0.5f*(g00 + g11);      // sum_t psd / m2

#pragma unroll
  for (int s = 0; s < NSRC; ++s) {
    const float mk = 1.f/(1.f + __expf(-W[s*NBINS + b]));
    const float m2 = mk*mk;
    const float sc = m2 / (EPS_F + m2*P);
    float2* R = Rbuf + (((size_t)nchunk*NBINS + b)*NSRC + s)*4;
    R[0] = make_float2(sc*g00, 0.f);
    R[1] = make_float2(sc*g01r,  sc*g01i);
    R[2] = make_float2(sc*g01r, -sc*g01i);
    R[3] = make_float2(sc*g11, 0.f);
  }
}

// ---------------- per-(t,b) Wiener gain apply --------------------------------
__global__ void kapply(const float2* __restrict__ mix, const float* __restrict__ W,
                       const float* __restrict__ maxbuf,
                       const float2* __restrict__ Rbuf, float2* __restrict__ ybuf) {
  const long long idx   = (long long)blockIdx.x*256 + threadIdx.x;
  const long long total = (long long)NBATCH*2*NWIN*NBINS;
  if (idx >= total) return;
  const int b      = (int)(idx % NBINS);
  const int t      = (int)((idx / NBINS) % NWIN);
  const int nchunk = (int)(idx / ((long long)NBINS*NWIN));
  const int n = nchunk >> 1, ch = nchunk & 1;
  const int fr = ch*NWIN + t;

  const float ma  = fmaxf(1.f, sqrtf(maxbuf[nchunk]) * 0.1f);
  const float ima = 1.f/ma;
  float2 x0 = mix[(((size_t)(n*NCH + 0))*NFRAMES + fr)*NBINS + b];
  float2 x1 = mix[(((size_t)(n*NCH + 1))*NFRAMES + fr)*NBINS + b];
  x0 = cscale(x0, ima); x1 = cscale(x1, ima);
  const float xx = 0.5f*(x0.x*x0.x + x0.y*x0.y + x1.x*x1.x + x1.y*x1.y);

  float  psd[NSRC];
  float2 R[NSRC][4];
  float2 C00 = make_float2(SQEPS_F, 0.f), C01 = make_float2(0.f, 0.f);
  float2 C10 = make_float2(0.f, 0.f),     C11 = make_float2(SQEPS_F, 0.f);
#pragma unroll
  for (int s = 0; s < NSRC; ++s) {
    const float mk = 1.f/(1.f + __expf(-W[s*NBINS + b]));
    psd[s] = mk*mk*xx;
    const float2* Rp = Rbuf + (((size_t)nchunk*NBINS + b)*NSRC + s)*4;
    R[s][0] = Rp[0]; R[s][1] = Rp[1]; R[s][2] = Rp[2]; R[s][3] = Rp[3];
    C00 = cadd(C00, cscale(R[s][0], psd[s]));
    C01 = cadd(C01, cscale(R[s][1], psd[s]));
    C10 = cadd(C10, cscale(R[s][2], psd[s]));
    C11 = cadd(C11, cscale(R[s][3], psd[s]));
  }
  // 2x2 complex inverse
  const float2 det  = csub(cmul(C00, C11), cmul(C01, C10));
  const float  dmag = det.x*det.x + det.y*det.y;
  const float2 idet = make_float2(det.x/dmag, -det.y/dmag);
  const float2 i00 = cmul(C11, idet);
  const float2 i01 = cmul(make_float2(-C01.x, -C01.y), idet);
  const float2 i10 = cmul(make_float2(-C10.x, -C10.y), idet);
  const float2 i11 = cmul(C00, idet);

#pragma unroll
  for (int s = 0; s < NSRC; ++s) {
    const float2 g00 = cscale(cadd(cmul(R[s][0], i00), cmul(R[s][1], i10)), psd[s]);
    const float2 g01 = cscale(cadd(cmul(R[s][0], i01), cmul(R[s][1], i11)), psd[s]);
    const float2 g10 = cscale(cadd(cmul(R[s][2], i00), cmul(R[s][3], i10)), psd[s]);
    const float2 g11 = cscale(cadd(cmul(R[s][2], i01), cmul(R[s][3], i11)), psd[s]);
    float2 y0 = cadd(cmul(g00, x0), cmul(g01, x1));
    float2 y1 = cadd(cmul(g10, x0), cmul(g11, x1));
    y0 = cscale(y0, ma); y1 = cscale(y1, ma);
    ybuf[((((size_t)(n*NSRC + s))*NCH + 0)*NFRAMES + fr)*NBINS + b] = y0;
    ybuf[((((size_t)(n*NSRC + s))*NCH + 1)*NFRAMES + fr)*NBINS + b] = y1;
  }
}

// ---------------- overlap-add + window-sum normalize + crop ------------------
__global__ void kola(const float* __restrict__ frames, float* __restrict__ out) {
  const long long idx   = (long long)blockIdx.x*256 + threadIdx.x;
  const long long total = (long long)NBATCH*NSRC*NCH*TLEN;
  if (idx >= total) return;
  const int t   = (int)(idx % TLEN);
  const int nsc = (int)(idx / TLEN);
  const int pos = t + NFFT/2;                  // padded coordinates
  int f1 = pos >> 10; if (f1 > NFRAMES - 1) f1 = NFRAMES - 1;
  const int f0 = (pos >= NFFT) ? ((pos - (NFFT - 1) + (HOP - 1)) >> 10) : 0;

  float acc = 0.f, wsum = 0.f;
  for (int f = f0; f <= f1; ++f) {
    const int off = pos - f*HOP;
    acc += frames[((size_t)nsc*NFRAMES + f)*NFFT + off];
    const float w = hann_w(off);
    wsum += w*w;
  }
  out[idx] = acc / (wsum > 1e-11f ? wsum : 1.f);
}

// ---------------- launch ------------------------------------------------------
extern "C" void kernel_launch(void* const* d_in, const int* in_sizes, int n_in,
                              void* d_out, int out_size, void* d_ws, size_t ws_size,
                              hipStream_t stream) {
  (void)in_sizes; (void)n_in; (void)out_size; (void)ws_size;
  const float* audio = (const float*)d_in[0];
  const float* W     = (const float*)d_in[1];
  // d_in[2] is niter; the workload fixes it to 1, implemented as one EM pass
  // (the y=mask*x initialization makes the single-pass statistics closed-form).

  char* ws = (char*)d_ws;
  size_t off = 0;
  auto carve = [&](size_t bytes) -> void* {
    void* p = ws + off;
    off += (bytes + 255) & ~(size_t)255;
    return p;
  };
  float2* mix    = (float2*)carve((size_t)NBATCH*NCH*NFRAMES*NBINS*sizeof(float2));
  float2* ybuf   = (float2*)carve((size_t)NBATCH*NSRC*NCH*NFRAMES*NBINS*sizeof(float2));
  float*  frames = (float*) carve((size_t)NBATCH*NSRC*NCH*NFRAMES*NFFT*sizeof(float));
  float2* Rbuf   = (float2*)carve((size_t)NBATCH*2*NBINS*NSRC*4*sizeof(float2));
  float*  maxbuf = (float*) carve(NBATCH*2*sizeof(float));

  kinit<<<dim3(1), dim3(64), 0, stream>>>(maxbuf);

  kfft_fwd<<<dim3(NBATCH*NCH*NFRAMES), dim3(128), 0, stream>>>(audio, mix);

  {
    const int per = NCH*NWIN*NBINS;
    kmax<<<dim3((per + 255)/256, NBATCH*2), dim3(256), 0, stream>>>(mix, maxbuf);
  }
  kstats<<<dim3((NBINS + 255)/256, NBATCH*2), dim3(256), 0, stream>>>(mix, W, maxbuf, Rbuf);
  {
    const long long total = (long long)NBATCH*2*NWIN*NBINS;
    kapply<<<dim3((unsigned)((total + 255)/256)), dim3(256), 0, stream>>>(
        mix, W, maxbuf, Rbuf, ybuf);
  }

  kfft_inv<<<dim3(NBATCH*NSRC*NCH*NFRAMES), dim3(128), 0, stream>>>(ybuf, frames);

  {
    const long long total = (long long)NBATCH*NSRC*NCH*TLEN;
    kola<<<dim3((unsigned)((total + 255)/256)), dim3(256), 0, stream>>>(
        frames, (float*)d_out);
  }
}